// AEDecoder_66340064854754
// MI455X (gfx1250) — compile-verified
//
#include <hip/hip_runtime.h>

typedef float v2f __attribute__((ext_vector_type(2)));
typedef float v8f __attribute__((ext_vector_type(8)));

#define T_   512
#define K_   4
#define G_   20000
#define EPG_ 3
#define B_   512
#define BT   16     // b-tile per block (one WMMA M dimension)
#define GT   512    // g-tile per block
#define NW   16     // waves per block (512 threads, wave32)

__device__ __forceinline__ float lrelu(float x) {
    return x > 0.f ? x : 0.01f * x;
}

__global__ __launch_bounds__(512)
void ae_decoder_fused(const float* __restrict__ features,
                      const float* __restrict__ w1,
                      const float* __restrict__ b1,
                      const float* __restrict__ w2,
                      const float* __restrict__ b2,
                      const float* __restrict__ w3,
                      const float* __restrict__ b3,
                      const int*   __restrict__ edge_tf,
                      float*       __restrict__ out)
{
    // h2 tile for 16 b's over all t: 16*512*4 f32 = 128 KB (WGP LDS is 320 KB)
    __shared__ __align__(16) float lds_h2[BT * T_ * K_];

    const int tid  = threadIdx.x;
    const int wave = tid >> 5;
    const int lane = tid & 31;
    const int half = lane >> 4;   // 0: lanes 0-15, 1: lanes 16-31
    const int m    = lane & 15;

    const int bbase = blockIdx.y * BT;
    const int gbase = blockIdx.x * GT;

    // ---------------- Phase A: stages 1+2 via V_WMMA_F32_16X16X4_F32 -------------
    // Per t: D(16x16) = A(16x4) x B(4x16) + C
    //   A[b,j] = lrelu(f[b,t]*w1[t,j] + b1[t,j])   (lane L: M=L%16, K=(L/16)*2+v)
    //   B[j,k] = w2[t,j,k] for k<4, else 0         (mirrored K layout)
    //   C[b,k] = b2[t,k]  (bias fused into accumulate)
    // D columns k<4 hold pre-activation h2; apply lrelu and stash in LDS.
    const int j0 = half * 2;
    for (int i = 0; i < T_ / NW; ++i) {
        const int t = wave * (T_ / NW) + i;

        const float f = features[(size_t)(bbase + m) * T_ + t];
        v2f a;
        a.x = lrelu(f * w1[t * K_ + j0]     + b1[t * K_ + j0]);
        a.y = lrelu(f * w1[t * K_ + j0 + 1] + b1[t * K_ + j0 + 1]);

        float bx = 0.f, by = 0.f, cv = 0.f;
        if (m < K_) {
            bx = w2[(t * K_ + j0)     * K_ + m];
            by = w2[(t * K_ + j0 + 1) * K_ + m];
            cv = b2[t * K_ + m];
        }
        v2f bmat; bmat.x = bx; bmat.y = by;
        v8f c = {cv, cv, cv, cv, cv, cv, cv, cv};

        v8f d = __builtin_amdgcn_wmma_f32_16x16x4_f32(
            /*neg_a=*/false, a, /*neg_b=*/false, bmat,
            /*c_mod=*/(short)0, c, /*reuse_a=*/false, /*reuse_b=*/false);

        if (m < K_) {
            #pragma unroll
            for (int v = 0; v < 8; ++v) {
                const int brow = v + 8 * half;        // D row M -> local b
                lds_h2[(brow * T_ + t) * K_ + m] = lrelu(d[v]);
            }
        }
    }

    __syncthreads();

    // ---------------- Phase B: gather + length-12 dot, coalesced store -----------
    // wave w owns local b-row w; lanes sweep consecutive g -> coalesced 41 MB store.
    const int    bl    = wave;
    const size_t bglob = (size_t)(bbase + bl);
    const float4* __restrict__ w3v = (const float4*)w3;   // (g*EPG+e) -> 16B aligned

    for (int it = 0; it < GT / 32; ++it) {
        const int g = gbase + it * 32 + lane;
        if (g < G_) {
            float acc = b3[g];
            #pragma unroll
            for (int e = 0; e < EPG_; ++e) {
                const int t = edge_tf[g * EPG_ + e];
                const float4 h  = *(const float4*)&lds_h2[(bl * T_ + t) * K_];
                const float4 wv = w3v[g * EPG_ + e];
                acc = fmaf(h.x, wv.x, acc);
                acc = fmaf(h.y, wv.y, acc);
                acc = fmaf(h.z, wv.z, acc);
                acc = fmaf(h.w, wv.w, acc);
            }
            out[bglob * G_ + g] = acc;
        }
    }
}

extern "C" void kernel_launch(void* const* d_in, const int* in_sizes, int n_in,
                              void* d_out, int out_size, void* d_ws, size_t ws_size,
                              hipStream_t stream)
{
    const float* features = (const float*)d_in[0];
    const float* w1       = (const float*)d_in[1];
    const float* b1       = (const float*)d_in[2];
    const float* w2       = (const float*)d_in[3];
    const float* b2       = (const float*)d_in[4];
    const float* w3       = (const float*)d_in[5];
    const float* b3       = (const float*)d_in[6];
    const int*   edge_tf  = (const int*)d_in[7];
    float*       out      = (float*)d_out;

    dim3 grid((G_ + GT - 1) / GT, B_ / BT);   // 40 x 32 = 1280 blocks
    ae_decoder_fused<<<grid, 512, 0, stream>>>(features, w1, b1, w2, b2,
                                               w3, b3, edge_tf, out);
}